// TextureAnalysisModule_43997644980699
// MI455X (gfx1250) — compile-verified
//
#include <hip/hip_runtime.h>

typedef __attribute__((ext_vector_type(2))) float v2f;
typedef __attribute__((ext_vector_type(8))) float v8f;

constexpr int BATCH = 16, CH = 3, HW = 1024, PATCH = 32, NPATCH = 1024;
constexpr int KSEL = 9, G96 = 96, OUT = 224;

// ---------------- Stage 1: per-patch gradient statistics (bandwidth bound) ----
// One block per patch; 8 waves x 6 (b,c) slices each; lane = column in patch.
__global__ void stats_kernel(const float* __restrict__ x, float* __restrict__ scores) {
    int pid = blockIdx.x;
    int ph = pid >> 5, pw = pid & 31;
    int lane = threadIdx.x & 31;
    int wid  = threadIdx.x >> 5;

    float sdx = 0.f, qdx = 0.f, sdy = 0.f, qdy = 0.f;
    for (int pair = wid; pair < BATCH * CH; pair += 8) {
        const float* base = x + ((size_t)pair * HW + (size_t)ph * PATCH) * HW + (size_t)pw * PATCH;
        float prev = 0.f;
        for (int r = 0; r < PATCH; ++r) {
            float v  = base[(size_t)r * HW + lane];
            float vn = __shfl_down(v, 1);
            if (lane < 31) { float d = fabsf(vn - v); sdx += d; qdx += d * d; }
            if (r > 0)     { float d = fabsf(v - prev); sdy += d; qdy += d * d; }
            prev = v;
        }
    }
    // promote to f64 for the reduction + variance (sumsq - sum^2/n cancellation)
    double a = sdx, b = qdx, c = sdy, d = qdy;
    for (int off = 16; off > 0; off >>= 1) {
        a += __shfl_down(a, off);
        b += __shfl_down(b, off);
        c += __shfl_down(c, off);
        d += __shfl_down(d, off);
    }
    __shared__ double red[8][4];
    if (lane == 0) { red[wid][0] = a; red[wid][1] = b; red[wid][2] = c; red[wid][3] = d; }
    __syncthreads();
    if (threadIdx.x == 0) {
        double A = 0, B = 0, C = 0, D = 0;
        for (int w = 0; w < 8; ++w) { A += red[w][0]; B += red[w][1]; C += red[w][2]; D += red[w][3]; }
        const double n = (double)(BATCH * CH * PATCH * (PATCH - 1)); // 47616
        double mean = 0.5 * (A + C) / n;
        double var  = 0.5 * ((B - A * A / n) + (D - C * C / n)) / (n - 1.0);
        scores[pid] = (float)(mean * var);
    }
}

// ---------------- Stage 2: stable rank == argsort selection ------------------
// sel[0..8]  = rich (ranks 1015..1023, ascending score order)
// sel[16..24] = poor (ranks 0..8)
__global__ void rank_kernel(const float* __restrict__ scores, int* __restrict__ sel) {
    __shared__ float s[NPATCH];
    int i = threadIdx.x;
    s[i] = scores[i];
    __syncthreads();
    float si = s[i];
    int rank = 0;
    for (int j = 0; j < NPATCH; ++j) {
        float sj = s[j];
        rank += (sj < si) || (sj == si && j < i);
    }
    if (rank >= NPATCH - KSEL) sel[rank - (NPATCH - KSEL)] = i;
    if (rank < KSEL)           sel[16 + rank] = i;
}

// ---------------- Stage 3: gather selected patches into 96x96 mosaics --------
__global__ void gather_kernel(const float* __restrict__ x, const int* __restrict__ sel,
                              float* __restrict__ img96) {
    int idx = blockIdx.x * blockDim.x + threadIdx.x;
    const int total = 2 * BATCH * CH * G96 * G96;
    if (idx >= total) return;
    int t = idx;
    int xc = t % G96; t /= G96;
    int yc = t % G96; t /= G96;
    int c  = t % CH;  t /= CH;
    int b  = t % BATCH; t /= BATCH;
    int s  = t;                              // 0 = rich, 1 = poor
    int gi = yc >> 5, r = yc & 31;
    int gj = xc >> 5, col = xc & 31;
    int pid = sel[(s ? 16 : 0) + gi * 3 + gj];
    int ph = pid >> 5, pw = pid & 31;
    img96[idx] = x[(((size_t)(b * CH + c)) * HW + (size_t)ph * PATCH + r) * HW + (size_t)pw * PATCH + col];
}

// ---------------- Stage 4: dense bilinear interpolation matrix A[224][96] ----
// JAX half-pixel resize; edge-normalized triangle weights == clamp-to-edge.
__global__ void build_A(float* __restrict__ A) {
    int idx = blockIdx.x * blockDim.x + threadIdx.x;
    if (idx >= OUT * G96) return;
    int i = idx / G96, j = idx % G96;
    float src = (i + 0.5f) * ((float)G96 / (float)OUT) - 0.5f;
    float ff = floorf(src);
    int f0 = (int)ff;
    float w1 = src - ff;
    int c0 = min(max(f0, 0), G96 - 1);
    int c1 = min(max(f0 + 1, 0), G96 - 1);
    float a = 0.f;
    if (j == c0) a += 1.f - w1;
    if (j == c1) a += w1;
    A[idx] = a;
}

// ---------------- Stage 5: upsample via WMMA f32 16x16x4 ---------------------
// Fragment layouts per CDNA5 ISA 7.12.2:
//   A frag: A[m][k] -> lane 16*(k>>1) + m, vgpr k&1
//   B frag: B[k][n] -> lane 16*(k>>1) + n, vgpr k&1
//   C/D   : D[m][n] -> vgpr m%8, lane n + 16*(m/8)
__global__ void upsample_rows_wmma(const float* __restrict__ A,
                                   const float* __restrict__ img96,
                                   float* __restrict__ T) {
    // T[img][224][96] = A(224x96) x img96[img](96x96); one wave per 16x16 tile
    int wave = (blockIdx.x * blockDim.x + threadIdx.x) >> 5;
    int lane = threadIdx.x & 31;
    const int tiles_n = 6, tiles_per_img = 14 * 6;
    int img = wave / tiles_per_img;
    int t   = wave % tiles_per_img;
    int tm = t / tiles_n, tn = t % tiles_n;
    if (img >= 2 * BATCH * CH) return;      // wave-uniform guard
    const float* B = img96 + (size_t)img * G96 * G96;
    float* outp = T + (size_t)img * OUT * G96;

    int m     = tm * 16 + (lane & 15);
    int n     = tn * 16 + (lane & 15);
    int klane = (lane >> 4) * 2;

    v8f acc = {};
    for (int kk = 0; kk < G96; kk += 4) {
        v2f af, bf;
        af.x = A[m * G96 + kk + klane];
        af.y = A[m * G96 + kk + klane + 1];
        bf.x = B[(kk + klane) * G96 + n];
        bf.y = B[(kk + klane + 1) * G96 + n];
        acc = __builtin_amdgcn_wmma_f32_16x16x4_f32(false, af, false, bf,
                                                    (short)0, acc, false, false);
    }
    int row0 = tm * 16 + (lane >> 4) * 8;
    int col  = tn * 16 + (lane & 15);
#pragma unroll
    for (int r = 0; r < 8; ++r)
        outp[(row0 + r) * G96 + col] = acc[r];
}

__global__ void upsample_cols_wmma(const float* __restrict__ A,
                                   const float* __restrict__ T,
                                   float* __restrict__ U) {
    // U[img][224][224] = T[img](224x96) x A^T(96x224); B[k][n] = A[n][k]
    int wave = (blockIdx.x * blockDim.x + threadIdx.x) >> 5;
    int lane = threadIdx.x & 31;
    const int tiles_n = 14, tiles_per_img = 14 * 14;
    int img = wave / tiles_per_img;
    int t   = wave % tiles_per_img;
    int tm = t / tiles_n, tn = t % tiles_n;
    if (img >= 2 * BATCH * CH) return;      // wave-uniform guard
    const float* Ti = T + (size_t)img * OUT * G96;
    float* outp = U + (size_t)img * OUT * OUT;

    int m     = tm * 16 + (lane & 15);
    int ncol  = tn * 16 + (lane & 15);
    int klane = (lane >> 4) * 2;

    v8f acc = {};
    for (int kk = 0; kk < G96; kk += 4) {
        v2f af, bf;
        af.x = Ti[m * G96 + kk + klane];
        af.y = Ti[m * G96 + kk + klane + 1];
        bf.x = A[ncol * G96 + kk + klane];
        bf.y = A[ncol * G96 + kk + klane + 1];
        acc = __builtin_amdgcn_wmma_f32_16x16x4_f32(false, af, false, bf,
                                                    (short)0, acc, false, false);
    }
    int row0 = tm * 16 + (lane >> 4) * 8;
    int col  = tn * 16 + (lane & 15);
#pragma unroll
    for (int r = 0; r < 8; ++r)
        outp[(row0 + r) * OUT + col] = acc[r];
}

// ---------------- Stage 6: depthwise 3x3 SRM stencil, zero padding -----------
__global__ void srm_conv(const float* __restrict__ U, const float* __restrict__ kflt,
                         float* __restrict__ out) {
    int idx = blockIdx.x * blockDim.x + threadIdx.x;
    const int total = 2 * BATCH * CH * OUT * OUT;
    if (idx >= total) return;
    int j = idx % OUT;
    int i = (idx / OUT) % OUT;
    int plane = idx / (OUT * OUT);
    int c = plane % CH;
    const float* in = U + (size_t)plane * OUT * OUT;
    const float* f  = kflt + 36 * c;   // k[c][c][:][:] flat offset = c*27 + c*9
    float acc = 0.f;
#pragma unroll
    for (int di = 0; di < 3; ++di) {
        int ii = i + di - 1;
        if (ii < 0 || ii >= OUT) continue;
#pragma unroll
        for (int dj = 0; dj < 3; ++dj) {
            int jj = j + dj - 1;
            if (jj < 0 || jj >= OUT) continue;
            acc += f[di * 3 + dj] * in[ii * OUT + jj];
        }
    }
    out[idx] = acc;
}

extern "C" void kernel_launch(void* const* d_in, const int* in_sizes, int n_in,
                              void* d_out, int out_size, void* d_ws, size_t ws_size,
                              hipStream_t stream) {
    const float* x    = (const float*)d_in[0];
    const float* kflt = (const float*)d_in[1];
    float* out = (float*)d_out;
    float* ws  = (float*)d_ws;

    float* scores = ws;                         // 1024 f
    int*   sel    = (int*)(ws + 1024);          // 32 ints (rich @0, poor @16)
    float* A      = ws + 2048;                  // 224*96   = 21504 f
    float* img96  = A + OUT * G96;              // 2*16*3*96*96   = 884736 f
    float* T      = img96 + 2 * BATCH * CH * G96 * G96;   // 2*16*3*224*96 = 2064384 f
    float* U      = T + 2 * BATCH * CH * OUT * G96;       // 2*16*3*224*224 = 4816896 f

    stats_kernel<<<NPATCH, 256, 0, stream>>>(x, scores);
    rank_kernel<<<1, NPATCH, 0, stream>>>(scores, sel);
    gather_kernel<<<(2 * BATCH * CH * G96 * G96 + 255) / 256, 256, 0, stream>>>(x, sel, img96);
    build_A<<<(OUT * G96 + 255) / 256, 256, 0, stream>>>(A);
    // 96 images x 84 tiles = 8064 waves = 1008 blocks of 8 waves (EXEC all-ones)
    upsample_rows_wmma<<<1008, 256, 0, stream>>>(A, img96, T);
    // 96 images x 196 tiles = 18816 waves = 2352 blocks
    upsample_cols_wmma<<<2352, 256, 0, stream>>>(A, T, U);
    srm_conv<<<(2 * BATCH * CH * OUT * OUT + 255) / 256, 256, 0, stream>>>(U, kflt, out);
}